// Mamba_83897891160356
// MI455X (gfx1250) — compile-verified
//
#include <hip/hip_runtime.h>
#include <hip/hip_bf16.h>
#include <math.h>

// ---------------------------------------------------------------------------
// Bidirectional Mamba block for MI455X (gfx1250), fp32 throughout.
// GEMMs use V_WMMA_F32_16X16X4_F32 (wave32 matrix core, exact fp32).
// Scan recomputes exp(dt*A) on the fly (v_exp_f32) -> no 256MB dA/dBx
// intermediates; total workspace ~81MB fits in the 192MB L2.
// ---------------------------------------------------------------------------

typedef __attribute__((ext_vector_type(2))) float v2f;
typedef __attribute__((ext_vector_type(8))) float v8f;

#define WMMA_F32_K4(A, B, C) \
  __builtin_amdgcn_wmma_f32_16x16x4_f32(false, (A), false, (B), (short)0, (C), false, false)

static constexpr int D_MODEL = 1024;
static constexpr int D_INNER = 2048;
static constexpr int D_STATE = 16;
static constexpr int DT_RANK = 64;
static constexpr int NPROJ   = DT_RANK + 2 * D_STATE; // 96
static constexpr int BATCH   = 2;
static constexpr int SEQLEN  = 1024;

// ---------------------------------------------------------------------------
// Kernel 1: in_proj   xz[b,e,l] = sum_d W[e,d] * H[b,l,d]
// W: [4096,1024] row-major;  H: [2,1024,1024] (b,l,d);  XZ: [2,4096,1024]
// Wave tile: M=16 (e) x N=64 (l), K stepped by 4.  A frag and B frag are both
// K-contiguous in memory -> single b64 loads per fragment.
// ---------------------------------------------------------------------------
__global__ void __launch_bounds__(256)
inproj_kernel(const float* __restrict__ W, const float* __restrict__ H,
              float* __restrict__ XZ) {
  const int lane = threadIdx.x & 31;
  const int wv   = threadIdx.x >> 5;
  const int r    = lane & 15;       // row (A) / col (B,D)
  const int hh   = lane >> 4;       // K-half select
  const int m0   = blockIdx.x * 128 + wv * 16;   // e tile
  const int l0   = blockIdx.y * 64;              // l tile
  const int b    = blockIdx.z;

  const float* arow = W + (size_t)(m0 + r) * D_MODEL + 2 * hh;
  const float* bcol = H + ((size_t)b * SEQLEN + l0 + r) * D_MODEL + 2 * hh;

  v8f acc0 = {}, acc1 = {}, acc2 = {}, acc3 = {};
  for (int k = 0; k < D_MODEL; k += 4) {
    v2f a  = *(const v2f*)(arow + k);
    v2f b0 = *(const v2f*)(bcol + k);
    v2f b1 = *(const v2f*)(bcol + (size_t)16 * D_MODEL + k);
    v2f b2 = *(const v2f*)(bcol + (size_t)32 * D_MODEL + k);
    v2f b3 = *(const v2f*)(bcol + (size_t)48 * D_MODEL + k);
    acc0 = WMMA_F32_K4(a, b0, acc0);
    acc1 = WMMA_F32_K4(a, b1, acc1);
    acc2 = WMMA_F32_K4(a, b2, acc2);
    acc3 = WMMA_F32_K4(a, b3, acc3);
  }
  float* o = XZ + ((size_t)b * 2 * D_INNER + m0 + 8 * hh) * SEQLEN + l0 + r;
#pragma unroll
  for (int v = 0; v < 8; ++v) {
    o[(size_t)v * SEQLEN +  0] = acc0[v];
    o[(size_t)v * SEQLEN + 16] = acc1[v];
    o[(size_t)v * SEQLEN + 32] = acc2[v];
    o[(size_t)v * SEQLEN + 48] = acc3[v];
  }
}

// ---------------------------------------------------------------------------
// Kernel 2: causal depthwise conv(k=4) + SiLU over logical time t.
// reverse=1 reads x time-reversed (backward direction); U stored in logical t.
// ---------------------------------------------------------------------------
__global__ void __launch_bounds__(256)
conv_silu_kernel(const float* __restrict__ XZ, const float* __restrict__ w,
                 const float* __restrict__ bias, float* __restrict__ U,
                 int reverse) {
  const int idx = blockIdx.x * 256 + threadIdx.x;     // 2*2048*1024
  const int t = idx & (SEQLEN - 1);
  const int d = (idx >> 10) & (D_INNER - 1);
  const int b = idx >> 21;
  const float* x = XZ + ((size_t)b * 2 * D_INNER + d) * SEQLEN;
  float acc = bias[d];
#pragma unroll
  for (int j = 0; j < 4; ++j) {
    const int tj = t - 3 + j;
    if (tj >= 0) {
      const int p = reverse ? (SEQLEN - 1 - tj) : tj;
      acc = fmaf(w[d * 4 + j], x[p], acc);
    }
  }
  U[((size_t)b * D_INNER + d) * SEQLEN + t] = acc / (1.f + __expf(-acc));
}

// ---------------------------------------------------------------------------
// Kernel 3: x_dbl[b,k,l] = sum_d xproj_w[k,d] * U[b,d,l]   (96 x 2048 x 1024)
// Wave tile M=16 x N=64; A K-contiguous (b64), B K-strided (2x b32).
// ---------------------------------------------------------------------------
__global__ void __launch_bounds__(256)
xdbl_kernel(const float* __restrict__ Wp, const float* __restrict__ U,
            float* __restrict__ XD) {
  const int lane = threadIdx.x & 31;
  const int wv   = threadIdx.x >> 5;
  const int r    = lane & 15;
  const int hh   = lane >> 4;
  const int flat = blockIdx.x * 8 + wv;           // 0..191
  const int m0   = (flat % 6) * 16;
  const int l0   = ((flat / 6) & 15) * 64;
  const int b    = flat / 96;

  const float* arow = Wp + (size_t)(m0 + r) * D_INNER + 2 * hh;
  const float* bp   = U + ((size_t)b * D_INNER + 2 * hh) * SEQLEN + l0 + r;

  v8f acc0 = {}, acc1 = {}, acc2 = {}, acc3 = {};
  for (int k = 0; k < D_INNER; k += 4) {
    v2f a = *(const v2f*)(arow + k);
    const float* q = bp + (size_t)k * SEQLEN;
    v2f b0, b1, b2, b3;
    b0.x = q[0];  b0.y = q[SEQLEN];
    b1.x = q[16]; b1.y = q[SEQLEN + 16];
    b2.x = q[32]; b2.y = q[SEQLEN + 32];
    b3.x = q[48]; b3.y = q[SEQLEN + 48];
    acc0 = WMMA_F32_K4(a, b0, acc0);
    acc1 = WMMA_F32_K4(a, b1, acc1);
    acc2 = WMMA_F32_K4(a, b2, acc2);
    acc3 = WMMA_F32_K4(a, b3, acc3);
  }
  float* o = XD + ((size_t)b * NPROJ + m0 + 8 * hh) * SEQLEN + l0 + r;
#pragma unroll
  for (int v = 0; v < 8; ++v) {
    o[(size_t)v * SEQLEN +  0] = acc0[v];
    o[(size_t)v * SEQLEN + 16] = acc1[v];
    o[(size_t)v * SEQLEN + 32] = acc2[v];
    o[(size_t)v * SEQLEN + 48] = acc3[v];
  }
}

// ---------------------------------------------------------------------------
// Kernel 4: delta[b,d,l] = softplus(sum_r dtw[d,r]*XD[b,r,l] + dt_bias[d])
// M=2048, K=64 (16 WMMA steps), N=1024.
// ---------------------------------------------------------------------------
__global__ void __launch_bounds__(256)
delta_kernel(const float* __restrict__ dtw, const float* __restrict__ XD,
             const float* __restrict__ dtb, float* __restrict__ DL) {
  const int lane = threadIdx.x & 31;
  const int wv   = threadIdx.x >> 5;
  const int r    = lane & 15;
  const int hh   = lane >> 4;
  const int flat = blockIdx.x * 8 + wv;             // 0..4095
  const int m0   = (flat & 127) * 16;
  const int l0   = ((flat >> 7) & 15) * 64;
  const int b    = flat >> 11;

  const float* arow = dtw + (size_t)(m0 + r) * DT_RANK + 2 * hh;
  const float* bp   = XD + ((size_t)b * NPROJ + 2 * hh) * SEQLEN + l0 + r;

  v8f acc0 = {}, acc1 = {}, acc2 = {}, acc3 = {};
#pragma unroll
  for (int k = 0; k < DT_RANK; k += 4) {
    v2f a = *(const v2f*)(arow + k);
    const float* q = bp + (size_t)k * SEQLEN;
    v2f b0, b1, b2, b3;
    b0.x = q[0];  b0.y = q[SEQLEN];
    b1.x = q[16]; b1.y = q[SEQLEN + 16];
    b2.x = q[32]; b2.y = q[SEQLEN + 32];
    b3.x = q[48]; b3.y = q[SEQLEN + 48];
    acc0 = WMMA_F32_K4(a, b0, acc0);
    acc1 = WMMA_F32_K4(a, b1, acc1);
    acc2 = WMMA_F32_K4(a, b2, acc2);
    acc3 = WMMA_F32_K4(a, b3, acc3);
  }
  float* o = DL + ((size_t)b * D_INNER + m0 + 8 * hh) * SEQLEN + l0 + r;
#pragma unroll
  for (int v = 0; v < 8; ++v) {
    const float bias = dtb[m0 + 8 * hh + v];
    float x0 = acc0[v] + bias, x1 = acc1[v] + bias;
    float x2 = acc2[v] + bias, x3 = acc3[v] + bias;
    o[(size_t)v * SEQLEN +  0] = (x0 > 20.f) ? x0 : log1pf(__expf(x0));
    o[(size_t)v * SEQLEN + 16] = (x1 > 20.f) ? x1 : log1pf(__expf(x1));
    o[(size_t)v * SEQLEN + 32] = (x2 > 20.f) ? x2 : log1pf(__expf(x2));
    o[(size_t)v * SEQLEN + 48] = (x3 > 20.f) ? x3 : log1pf(__expf(x3));
  }
}

// ---------------------------------------------------------------------------
// Kernel 5: selective scan + skip + SiLU(z) gate.
// Lane layout: n = lane&15 (state dim), 2 d-channels per wave.
// h_t = exp(dt*A)*h_{t-1} + dt*B_t*u_t ; y_t = sum_n h*C + D*u ; out = y*silu(z)
// Backward dir (reverse=1) consumes reversed-time u/delta/xdbl and scatters the
// gated output back to original time, accumulating into Y.
// ---------------------------------------------------------------------------
__global__ void __launch_bounds__(256)
scan_kernel(const float* __restrict__ DL, const float* __restrict__ U,
            const float* __restrict__ XD, const float* __restrict__ A_log,
            const float* __restrict__ Dp, const float* __restrict__ XZ,
            float* __restrict__ Y, int reverse, int accumulate) {
  const int lane = threadIdx.x & 31;
  const int wv   = threadIdx.x >> 5;
  const int n    = lane & 15;
  const int ds   = lane >> 4;
  const int flat = blockIdx.x * 8 + wv;             // 0..2047
  const int b    = flat >> 10;
  const int d    = ((flat & 1023) << 1) + ds;

  const float A     = -__expf(A_log[d * D_STATE + n]);
  const float Dskip = Dp[d];
  const float* dl = DL + ((size_t)b * D_INNER + d) * SEQLEN;
  const float* ul = U  + ((size_t)b * D_INNER + d) * SEQLEN;
  const float* Bl = XD + ((size_t)b * NPROJ + DT_RANK + n) * SEQLEN;
  const float* Cl = XD + ((size_t)b * NPROJ + DT_RANK + D_STATE + n) * SEQLEN;
  const float* zl = XZ + ((size_t)b * 2 * D_INNER + D_INNER + d) * SEQLEN;
  float* yl = Y + ((size_t)b * D_INNER + d) * SEQLEN;

  float h = 0.f;
  for (int t = 0; t < SEQLEN; ++t) {
    __builtin_prefetch(Bl + t + 128, 0, 1);
    __builtin_prefetch(Cl + t + 128, 0, 1);
    const float dv = dl[t];
    const float uv = ul[t];
    const float dA = __expf(dv * A);
    h = fmaf(dA, h, dv * Bl[t] * uv);
    float p = h * Cl[t];
    p += __shfl_xor(p, 1, 32);
    p += __shfl_xor(p, 2, 32);
    p += __shfl_xor(p, 4, 32);
    p += __shfl_xor(p, 8, 32);
    if (n == 0) {
      const int lo = reverse ? (SEQLEN - 1 - t) : t;
      const float z = zl[lo];
      const float val = (p + Dskip * uv) * (z / (1.f + __expf(-z)));
      if (accumulate) yl[lo] += val; else yl[lo] = val;
    }
  }
}

// ---------------------------------------------------------------------------
// Kernel 6: out[b,l,o] = sum_d Wo[o,d] * Y[b,d,l]
// M=o (A K-contiguous b64), N=l (B K-strided), K=2048.
// ---------------------------------------------------------------------------
__global__ void __launch_bounds__(256)
outproj_kernel(const float* __restrict__ Wo, const float* __restrict__ Y,
               float* __restrict__ OUT) {
  const int lane = threadIdx.x & 31;
  const int wv   = threadIdx.x >> 5;
  const int r    = lane & 15;
  const int hh   = lane >> 4;
  const int flat = blockIdx.x * 8 + wv;             // 0..2047
  const int o0   = (flat & 63) * 16;
  const int l0   = ((flat >> 6) & 15) * 64;
  const int b    = flat >> 10;

  const float* arow = Wo + (size_t)(o0 + r) * D_INNER + 2 * hh;
  const float* bp   = Y + ((size_t)b * D_INNER + 2 * hh) * SEQLEN + l0 + r;

  v8f acc0 = {}, acc1 = {}, acc2 = {}, acc3 = {};
  for (int k = 0; k < D_INNER; k += 4) {
    v2f a = *(const v2f*)(arow + k);
    const float* q = bp + (size_t)k * SEQLEN;
    v2f b0, b1, b2, b3;
    b0.x = q[0];  b0.y = q[SEQLEN];
    b1.x = q[16]; b1.y = q[SEQLEN + 16];
    b2.x = q[32]; b2.y = q[SEQLEN + 32];
    b3.x = q[48]; b3.y = q[SEQLEN + 48];
    acc0 = WMMA_F32_K4(a, b0, acc0);
    acc1 = WMMA_F32_K4(a, b1, acc1);
    acc2 = WMMA_F32_K4(a, b2, acc2);
    acc3 = WMMA_F32_K4(a, b3, acc3);
  }
  // D element (m=o0+v+8h, n=l0+j*16+r) -> OUT[b, l, o]
  float* o = OUT + ((size_t)b * SEQLEN + l0 + r) * D_MODEL + o0 + 8 * hh;
#pragma unroll
  for (int v = 0; v < 8; ++v) {
    o[(size_t)( 0) * D_MODEL + v] = acc0[v];
    o[(size_t)(16) * D_MODEL + v] = acc1[v];
    o[(size_t)(32) * D_MODEL + v] = acc2[v];
    o[(size_t)(48) * D_MODEL + v] = acc3[v];
  }
}

// ---------------------------------------------------------------------------
extern "C" void kernel_launch(void* const* d_in, const int* in_sizes, int n_in,
                              void* d_out, int out_size, void* d_ws, size_t ws_size,
                              hipStream_t stream) {
  const float* hidden     = (const float*)d_in[0];
  const float* in_proj_w  = (const float*)d_in[1];
  const float* conv_w     = (const float*)d_in[2];
  const float* conv_b     = (const float*)d_in[3];
  const float* xproj_w    = (const float*)d_in[4];
  const float* dtproj_w   = (const float*)d_in[5];
  const float* dt_bias    = (const float*)d_in[6];
  const float* A_log      = (const float*)d_in[7];
  const float* D_skip     = (const float*)d_in[8];
  const float* convb_w    = (const float*)d_in[9];
  const float* convb_b    = (const float*)d_in[10];
  const float* xproj_b_w  = (const float*)d_in[11];
  const float* dtproj_b_w = (const float*)d_in[12];
  const float* dt_b_bias  = (const float*)d_in[13];
  const float* A_b_log    = (const float*)d_in[14];
  const float* D_b_skip   = (const float*)d_in[15];
  const float* out_proj_w = (const float*)d_in[16];
  float* out = (float*)d_out;

  // Workspace layout (floats). Reused between directions. ~81 MB total.
  float* xz   = (float*)d_ws;                                  // [2,4096,1024]
  float* u    = xz   + (size_t)BATCH * 2 * D_INNER * SEQLEN;   // [2,2048,1024]
  float* xdbl = u    + (size_t)BATCH * D_INNER * SEQLEN;       // [2,96,1024]
  float* dl   = xdbl + (size_t)BATCH * NPROJ * SEQLEN;         // [2,2048,1024]
  float* yacc = dl   + (size_t)BATCH * D_INNER * SEQLEN;       // [2,2048,1024]

  inproj_kernel<<<dim3(32, 16, 2), 256, 0, stream>>>(in_proj_w, hidden, xz);

  // Forward direction (writes yacc)
  conv_silu_kernel<<<16384, 256, 0, stream>>>(xz, conv_w, conv_b, u, 0);
  xdbl_kernel<<<24, 256, 0, stream>>>(xproj_w, u, xdbl);
  delta_kernel<<<512, 256, 0, stream>>>(dtproj_w, xdbl, dt_bias, dl);
  scan_kernel<<<256, 256, 0, stream>>>(dl, u, xdbl, A_log, D_skip, xz, yacc, 0, 0);

  // Backward direction (accumulates into yacc at reversed time)
  conv_silu_kernel<<<16384, 256, 0, stream>>>(xz, convb_w, convb_b, u, 1);
  xdbl_kernel<<<24, 256, 0, stream>>>(xproj_b_w, u, xdbl);
  delta_kernel<<<512, 256, 0, stream>>>(dtproj_b_w, xdbl, dt_b_bias, dl);
  scan_kernel<<<256, 256, 0, stream>>>(dl, u, xdbl, A_b_log, D_b_skip, xz, yacc, 1, 1);

  outproj_kernel<<<256, 256, 0, stream>>>(out_proj_w, yacc, out);
}